// MultiParamAttentionInterpolator_12223476924816
// MI455X (gfx1250) — compile-verified
//
#include <hip/hip_runtime.h>
#include <hip/hip_bf16.h>
#include <stdint.h>

#ifndef __has_builtin
#define __has_builtin(x) 0
#endif

#define M_  32
#define T_  50
#define NX_ 192
#define NY_ 192
#define NG_ 50
#define H_  4
#define D_  8
#define LX_ 100.0f
#define SIGMA_ 0.2f

// workspace layout (byte offsets)
#define WS_W   0              // float[32]   combined weights
#define WS_XI  128            // int[64]     x lower index per px
#define WS_XA  384            // float[64]   (1-tx)*insideX
#define WS_XB  640            // float[64]   tx*insideX
#define WS_YJ  896            // int[32*64]  y lower index per (m,py)
#define WS_YA  (896+8192)     // float[32*64] (1-ty)*insideY
#define WS_YB  (896+16384)    // float[32*64] ty*insideY

typedef __attribute__((ext_vector_type(2))) float v2f;
typedef __attribute__((ext_vector_type(8))) float v8f;
typedef int v4i_ __attribute__((vector_size(16)));

// ---------------- CDNA5 async global->LDS copy helpers ----------------
__device__ __forceinline__ void async_b128(const void* g, void* l) {
#if __has_builtin(__builtin_amdgcn_global_load_async_to_lds_b128)
  // clang-22 signature (from diagnostic): (v4i addrspace(1)*, v4i addrspace(3)*, imm, imm)
  // generic->AS1 keeps the 64-bit value; generic LDS pointer's low 32 bits are
  // the LDS offset -> AS3 pointer.
  __builtin_amdgcn_global_load_async_to_lds_b128(
      (__attribute__((address_space(1))) v4i_*)(uintptr_t)g,
      (__attribute__((address_space(3))) v4i_*)(uintptr_t)(uint32_t)(uintptr_t)l,
      0, 0);
#else
  *(float4*)l = *(const float4*)g;   // synchronous fallback
#endif
}

__device__ __forceinline__ void async_wait0() {
#if __has_builtin(__builtin_amdgcn_s_wait_asynccnt)
  __builtin_amdgcn_s_wait_asynccnt(0);
#else
  __asm__ volatile("s_wait_asynccnt 0" ::: "memory");
#endif
}

// ---------------- setup: weights + interpolation coefficients ----------------
__global__ __launch_bounds__(256) void setup_kernel(
    const float* __restrict__ ps, const float* __restrict__ tp,
    const float* __restrict__ Wq, const float* __restrict__ Wk,
    const float* __restrict__ sxg, const float* __restrict__ syg,
    char* __restrict__ ws)
{
  __shared__ float ntgt[3];
  __shared__ float nsrc[M_][3];
  __shared__ float qv[H_ * D_];
  __shared__ float lg[M_][H_];
  __shared__ float attn_s[M_];
  __shared__ float swv[M_];
  const int tid = threadIdx.x;

  if (tid == 0) {
    ntgt[0] = (tp[0] - 30.f) / 90.f;
    ntgt[1] = tp[1] / 0.0029f;
    ntgt[2] = tp[2] / 0.0018f;
  }
  if (tid < M_) {
    nsrc[tid][0] = (ps[tid*3+0] - 30.f) / 90.f;
    nsrc[tid][1] = ps[tid*3+1] / 0.0029f;
    nsrc[tid][2] = ps[tid*3+2] / 0.0018f;
  }
  __syncthreads();
  if (tid < H_ * D_)
    qv[tid] = ntgt[0]*Wq[tid*3+0] + ntgt[1]*Wq[tid*3+1] + ntgt[2]*Wq[tid*3+2];
  __syncthreads();
  if (tid < M_) {
    const float inv_sqrt_d = 0.35355339059327373f;  // 1/sqrt(8)
    for (int h = 0; h < H_; ++h) {
      float acc = 0.f;
      for (int d = 0; d < D_; ++d) {
        const int e = h * D_ + d;
        const float ke = nsrc[tid][0]*Wk[e*3+0] + nsrc[tid][1]*Wk[e*3+1] +
                         nsrc[tid][2]*Wk[e*3+2];
        acc += ke * qv[e];
      }
      lg[tid][h] = acc * inv_sqrt_d;
    }
    float d2 = 0.f;
    for (int c = 0; c < 3; ++c) {
      const float z = (nsrc[tid][c] - ntgt[c]) / SIGMA_;
      d2 += z * z;
    }
    swv[tid] = expf(-0.5f * d2);
  }
  __syncthreads();
  if (tid == 0) {
    for (int h = 0; h < H_; ++h) {               // stable softmax over sources
      float mx = -1e30f;
      for (int m = 0; m < M_; ++m) mx = fmaxf(mx, lg[m][h]);
      float s = 0.f;
      for (int m = 0; m < M_; ++m) { float e = expf(lg[m][h]-mx); lg[m][h]=e; s+=e; }
      for (int m = 0; m < M_; ++m) lg[m][h] /= s;
    }
    float ssum = 0.f;
    for (int m = 0; m < M_; ++m) {
      attn_s[m] = 0.25f*(lg[m][0]+lg[m][1]+lg[m][2]+lg[m][3]);
      ssum += swv[m];
    }
    ssum += 1e-8f;
    float wsum = 0.f;
    float* w = (float*)(ws + WS_W);
    for (int m = 0; m < M_; ++m) { float v = attn_s[m]*(swv[m]/ssum); swv[m]=v; wsum+=v; }
    wsum += 1e-8f;
    for (int m = 0; m < M_; ++m) w[m] = swv[m] / wsum;
  }

  // x coefficients (all m share src_xgrid, use row 0); searchsorted side='right'
  if (tid < NG_) {
    const float* gx = sxg;
    const float qx = LX_ * (float)tid / 49.0f;
    int cnt = 0;
    for (int k = 0; k < NX_; ++k) cnt += (gx[k] <= qx);
    const int i = min(max(cnt - 1, 0), NX_ - 2);
    const float tx = (qx - gx[i]) / (gx[i+1] - gx[i]);
    const float insx = (qx >= gx[0] && qx <= gx[NX_-1]) ? 1.f : 0.f;
    ((int*)(ws + WS_XI))[tid] = i;
    ((float*)(ws + WS_XA))[tid] = (1.f - tx) * insx;
    ((float*)(ws + WS_XB))[tid] = tx * insx;
  }
  // y coefficients per (m, py) on the scaled source grid
  const float lyt = tp[0];
  for (int idx = tid; idx < M_ * NG_; idx += 256) {
    const int m = idx / NG_, py = idx - (idx / NG_) * NG_;
    const float scale = lyt / ps[m*3+0];
    const float* gy = syg + m * NY_;
    const float qy = lyt * (float)py / 49.0f;
    int cnt = 0;
    for (int k = 0; k < NY_; ++k) cnt += (gy[k] * scale <= qy);
    const int j = min(max(cnt - 1, 0), NY_ - 2);
    const float gj = gy[j] * scale, gj1 = gy[j+1] * scale;
    const float ty = (qy - gj) / (gj1 - gj);
    const float g0 = gy[0] * scale, gN = gy[NY_-1] * scale;
    const float insy = (qy >= g0 && qy <= gN) ? 1.f : 0.f;
    ((int*)(ws + WS_YJ))[m*64 + py] = j;
    ((float*)(ws + WS_YA))[m*64 + py] = (1.f - ty) * insy;
    ((float*)(ws + WS_YB))[m*64 + py] = ty * insy;
  }
}

// ---------------- main: async-staged rows + f32 WMMA reduction ----------------
// WG = (field, t, Mtile). 4 waves, wave n owns output N-tile n.
// H (16x192) = x-lerp of staged rows (w folded into ByT along with y-lerp).
// out_tile(16x64) = sum_m  H_m(16x192) x By_m(192x64) via V_WMMA_F32_16X16X4_F32.
#define HP  196                 // H row pitch (floats): banks (4r+k)%64 -> conflict-free
#define BPT 196                 // ByT row pitch (floats), rows indexed by N
#define ROWS_F  (32 * NY_)      // staged rows buffer
#define HS_F    (16 * HP)
#define BYT_F   (64 * BPT)
#define SMEM_F  (ROWS_F + HS_F + BYT_F)

__global__ __launch_bounds__(128) void interp_kernel(
    const float* __restrict__ c1f, const float* __restrict__ c2f,
    const char* __restrict__ ws, float* __restrict__ out)
{
  extern __shared__ float smem[];
  float* rows = smem;                    // [32][192] staged field rows
  float* Hs   = smem + ROWS_F;           // [16][HP]  x-interpolated tile
  float* ByT  = smem + ROWS_F + HS_F;    // [64][BPT] B^T: ByT[n][k] = By[k][n]

  __shared__ int   sxi[NG_];
  __shared__ float sxa[NG_], sxb[NG_];

  const float* wv = (const float*)(ws + WS_W);
  const int*   yj = (const int*)(ws + WS_YJ);
  const float* yA = (const float*)(ws + WS_YA);
  const float* yB = (const float*)(ws + WS_YB);

  const int tid = threadIdx.x;
  if (tid < NG_) {
    sxi[tid] = ((const int*)(ws + WS_XI))[tid];
    sxa[tid] = ((const float*)(ws + WS_XA))[tid];
    sxb[tid] = ((const float*)(ws + WS_XB))[tid];
  }
  // initial full zero of ByT (per-m updates only clear/set their own entries)
  for (int i = tid; i < BYT_F; i += 128) ByT[i] = 0.f;

  const int b   = blockIdx.x;
  const int fid = b / (T_ * 4);
  const int r   = b - fid * (T_ * 4);
  const int t   = r >> 2;
  const int Mt  = r & 3;
  const float* Fbase = (fid == 0) ? c1f : c2f;

  const int wave = tid >> 5, lane = tid & 31;
  const int l15 = lane & 15, khalf = (lane >> 4) << 1;
  const int Nt = wave;                    // 4 waves -> 4 N-tiles

  v8f acc = {0.f, 0.f, 0.f, 0.f, 0.f, 0.f, 0.f, 0.f};
  int jprev = 0;                          // previous scatter rows (tid<NG_)

  __syncthreads();                        // sxi/sxa/sxb + ByT zero ready

  auto issue = [&](int m) {               // async-stage needed rows of F[m,t]
    const float* F = Fbase + (size_t)(m * T_ + t) * NX_ * NY_;
    for (int c = tid; c < 32 * (NY_ / 4); c += 128) {
      const int slot = c / (NY_ / 4), wi = c - (c / (NY_ / 4)) * (NY_ / 4);
      const int px = Mt * 16 + (slot >> 1);
      if (px < NG_) {
        const int row = sxi[px] + (slot & 1);
        async_b128(F + (size_t)row * NY_ + wi * 4, rows + slot * NY_ + wi * 4);
      }
    }
  };

  issue(0);

  for (int m = 0; m < M_; ++m) {
    async_wait0();
    __syncthreads();                      // rows[m] visible; prev GEMM done

    // build H (x-lerp); invalid px rows -> 0
    for (int idx = tid; idx < 16 * NY_; idx += 128) {
      const int p16 = idx / NY_, y = idx - (idx / NY_) * NY_;
      const int px = Mt * 16 + p16;
      float v = 0.f;
      if (px < NG_)
        v = sxa[px] * rows[(p16 * 2) * NY_ + y] +
            sxb[px] * rows[(p16 * 2 + 1) * NY_ + y];
      Hs[p16 * HP + y] = v;
    }
    // update ByT: column py touched only by thread py -> no intra-phase races
    if (tid < NG_) {
      ByT[tid * BPT + jprev]     = 0.f;
      ByT[tid * BPT + jprev + 1] = 0.f;
      const float wm = wv[m];
      const int j = yj[m * 64 + tid];
      ByT[tid * BPT + j]     = wm * yA[m * 64 + tid];
      ByT[tid * BPT + j + 1] = wm * yB[m * 64 + tid];
      jprev = j;
    }
    __syncthreads();                      // H/ByT ready; rows buffer free

    if (m + 1 < M_) issue(m + 1);         // prefetch next m under the GEMM

    // A 16x4 f32: lanes0-15 M=l15,K={kb,kb+1}; lanes16-31 M=l15,K={kb+2,kb+3}
    // B 4x16 f32: lanes0-15 N=l15,K rows {kb,kb+1}; lanes16-31 K rows {kb+2,kb+3}
    const float* Ap = Hs + l15 * HP + khalf;
    const float* Bp = ByT + (Nt * 16 + l15) * BPT + khalf;
    for (int kb = 0; kb < NY_; kb += 4) {
      v2f a = *(const v2f*)(Ap + kb);
      v2f bb = *(const v2f*)(Bp + kb);
      acc = __builtin_amdgcn_wmma_f32_16x16x4_f32(
          false, a, false, bb, (short)0, acc, false, false);
    }
    __syncthreads();                      // done reading H/ByT
  }

  // C/D layout: VGPR r: lanes0-15 -> M=r, N=lane; lanes16-31 -> M=r+8, N=lane-16
  float* outF = out + (size_t)(fid * T_ + t) * NG_ * NG_;
  const int py = Nt * 16 + l15;
#pragma unroll
  for (int rr = 0; rr < 8; ++rr) {
    const int px = Mt * 16 + rr + (lane >> 4) * 8;
    if (px < NG_ && py < NG_) outF[px * NG_ + py] = acc[rr];
  }
}

// ---------------- boundary columns ----------------
__global__ void finalize_kernel(float* __restrict__ out, const float* __restrict__ tp) {
  const int i = blockIdx.x * 256 + threadIdx.x;
  if (i < 2 * T_ * NG_) {
    const int f = i / (T_ * NG_);
    const int r = i - f * (T_ * NG_);
    const int t = r / NG_, px = r - (r / NG_) * NG_;
    if (f == 0)
      out[(size_t)t * NG_ * NG_ + px * NG_ + 0] = tp[1];          // c1[:,:,0]=ccu
    else
      out[(size_t)(T_ + t) * NG_ * NG_ + px * NG_ + (NG_ - 1)] = tp[2];  // c2[:,:,-1]=cni
  }
}

extern "C" void kernel_launch(void* const* d_in, const int* in_sizes, int n_in,
                              void* d_out, int out_size, void* d_ws, size_t ws_size,
                              hipStream_t stream) {
  const float* ps  = (const float*)d_in[0];   // params_src [32,3]
  const float* tp  = (const float*)d_in[1];   // tgt_params [3]
  const float* c1f = (const float*)d_in[2];   // c1_fields [32,50,192,192]
  const float* c2f = (const float*)d_in[3];   // c2_fields
  const float* Wq  = (const float*)d_in[4];   // [32,3]
  const float* Wk  = (const float*)d_in[5];   // [32,3]
  const float* sxg = (const float*)d_in[6];   // [32,192]
  const float* syg = (const float*)d_in[7];   // [32,192]
  float* out = (float*)d_out;
  char*  ws  = (char*)d_ws;

  setup_kernel<<<1, 256, 0, stream>>>(ps, tp, Wq, Wk, sxg, syg, ws);
  const size_t smem = (size_t)SMEM_F * sizeof(float);   // ~87 KB of the 320 KB WGP LDS
  interp_kernel<<<2 * T_ * 4, 128, smem, stream>>>(c1f, c2f, ws, out);
  finalize_kernel<<<(2 * T_ * NG_ + 255) / 256, 256, 0, stream>>>(out, tp);
}